// ProbabilisticAttention_21165598835387
// MI455X (gfx1250) — compile-verified
//
#include <hip/hip_runtime.h>
#include <math.h>

#define S 512
#define DM 512
#define NH 8
#define DK 64
#define EPSF 1e-6f

typedef float v2f __attribute__((ext_vector_type(2)));
typedef float v8f __attribute__((ext_vector_type(8)));

#if defined(__has_builtin)
#if __has_builtin(__builtin_amdgcn_tensor_load_to_lds) && \
    __has_builtin(__builtin_amdgcn_s_wait_tensorcnt)
#define USE_TDM 1
#endif
#endif

#ifdef USE_TDM
typedef unsigned int u32x4 __attribute__((ext_vector_type(4)));
typedef int i32x4 __attribute__((ext_vector_type(4)));
typedef int i32x8 __attribute__((ext_vector_type(8)));

// TDM load of a 32(row) x 64(dword) f32 tile, row stride 512 floats in
// global, into LDS with 4-dword padding after every 64 dwords (=> padded
// LDS row stride of 68 floats, matching ck[32][68]).  D# layout per
// CDNA5 ISA.
__device__ __forceinline__ void tdm_load_tile_32x64(const float* gsrc,
                                                    void* lds_dst) {
  unsigned long long ga = (unsigned long long)(uintptr_t)gsrc;
  unsigned int lds = (unsigned int)(uintptr_t)lds_dst;  // low 32b = LDS offset
  u32x4 g0;
  g0[0] = 1u;                                        // count=1, user mode
  g0[1] = lds;                                       // lds_addr [63:32]
  g0[2] = (unsigned int)(ga & 0xFFFFFFFFu);          // global_addr lo
  g0[3] = (unsigned int)((ga >> 32) & 0x01FFFFFFu)   // global_addr hi (57b)
        | (2u << 30);                                // type = 2 ("image")
  i32x8 g1;
  g1[0] = (2 << 16)      // data_size = 4B
        | (1 << 20)      // pad_enable
        | (5 << 22)      // pad_interval: 2^(5+1) = 64 dwords
        | (3 << 25);     // pad_amount: 3+1 = 4 dwords
  g1[1] = (64 << 16);    // tensor_dim0[15:0] = 64 at bits [63:48]
  g1[2] = (32 << 16);    // tensor_dim0 hi = 0 ; tensor_dim1 lo = 32
  g1[3] = (64 << 16);    // tensor_dim1 hi = 0 ; tile_dim0 = 64
  g1[4] = 32;            // tile_dim1 = 32 ; tile_dim2 = 0
  g1[5] = 512;           // tensor_dim0_stride lo32 = 512 elements
  g1[6] = 0;             // stride hi ; tensor_dim1_stride lo
  g1[7] = 0;
  i32x4 z4 = {0, 0, 0, 0};
#if defined(__clang_major__) && __clang_major__ >= 23
  i32x8 z8 = {0, 0, 0, 0, 0, 0, 0, 0};
  __builtin_amdgcn_tensor_load_to_lds(g0, g1, z4, z4, z8, 0);
#else
  __builtin_amdgcn_tensor_load_to_lds(g0, g1, z4, z4, 0);
#endif
}
#endif  // USE_TDM

// ---------------------------------------------------------------------------
// Kernel A: Y[512,512] = X[512,512] @ W^T + bias, fp32 WMMA 16x16x4.
// Block 256 thr (8 waves). Block tile 64(M) x 128(N); wave tile 16(M) x 64(N)
// = 4 WMMA accumulators sharing one A fragment per K-step.
// ---------------------------------------------------------------------------
__global__ __launch_bounds__(256) void proj_gemm(
    const float* __restrict__ X, const float* __restrict__ Wt,
    const float* __restrict__ bias, float* __restrict__ Y) {
  const int tid = threadIdx.x;
  const int wave = tid >> 5, lane = tid & 31;
  const int lane16 = lane & 15, hi = lane >> 4;

  const int m0 = blockIdx.x * 64 + (wave & 3) * 16;
  const int n0 = blockIdx.y * 128 + (wave >> 2) * 64;

  const float* xrow = X + (size_t)(m0 + lane16) * DM + hi * 2;
  const float* wrow0 = Wt + (size_t)(n0 + 0 * 16 + lane16) * DM + hi * 2;
  const float* wrow1 = Wt + (size_t)(n0 + 1 * 16 + lane16) * DM + hi * 2;
  const float* wrow2 = Wt + (size_t)(n0 + 2 * 16 + lane16) * DM + hi * 2;
  const float* wrow3 = Wt + (size_t)(n0 + 3 * 16 + lane16) * DM + hi * 2;

  v8f c0 = {0.f,0.f,0.f,0.f,0.f,0.f,0.f,0.f};
  v8f c1 = c0, c2 = c0, c3 = c0;

#pragma unroll 4
  for (int k0 = 0; k0 < DM; k0 += 4) {
    v2f a  = *(const v2f*)(xrow + k0);
    v2f b0 = *(const v2f*)(wrow0 + k0);
    v2f b1 = *(const v2f*)(wrow1 + k0);
    v2f b2 = *(const v2f*)(wrow2 + k0);
    v2f b3 = *(const v2f*)(wrow3 + k0);
    c0 = __builtin_amdgcn_wmma_f32_16x16x4_f32(false, a, false, b0, (short)0, c0, false, false);
    c1 = __builtin_amdgcn_wmma_f32_16x16x4_f32(false, a, false, b1, (short)0, c1, false, false);
    c2 = __builtin_amdgcn_wmma_f32_16x16x4_f32(false, a, false, b2, (short)0, c2, false, false);
    c3 = __builtin_amdgcn_wmma_f32_16x16x4_f32(false, a, false, b3, (short)0, c3, false, false);
  }

  v8f acc[4] = {c0, c1, c2, c3};
#pragma unroll
  for (int t = 0; t < 4; ++t) {
    const int n = n0 + t * 16 + lane16;
    const float bn = bias[n];
#pragma unroll
    for (int r = 0; r < 8; ++r) {
      const int m = m0 + r + hi * 8;
      Y[(size_t)m * DM + n] = acc[t][r] + bn;
    }
  }
}

// ---------------------------------------------------------------------------
// Kernel B: in-place x -> x*x (+eps), plus per-(s,h) LOG2 sums (log2 units
// avoid the *ln2 fixup on every v_log_f32).  One wave per (s,h).
// ---------------------------------------------------------------------------
__global__ __launch_bounds__(256) void sigma_post(
    float* __restrict__ buf, float* __restrict__ logsum, int add_eps) {
  const int wave = threadIdx.x >> 5, lane = threadIdx.x & 31;
  const int p = blockIdx.x * 8 + wave;
  const int s = p >> 3, h = p & 7;
  const size_t base = (size_t)s * DM + h * DK;

  float acc = 0.f;
#pragma unroll
  for (int i = 0; i < 2; ++i) {
    const int d = lane + 32 * i;
    float x = buf[base + d];
    float y = x * x + (add_eps ? EPSF : 0.f);
    buf[base + d] = y;
    if (logsum) acc += __log2f(y);
  }
  if (logsum) {
#pragma unroll
    for (int off = 16; off >= 1; off >>= 1) acc += __shfl_xor(acc, off, 32);
    if (lane == 0) logsum[s * NH + h] = acc;
  }
}

// ---------------------------------------------------------------------------
// Kernel C: Bhattacharyya distance + causal mask + softmax, one (h, 16-q
// tile) per block.  All log math in base 2; scores stored pre-scaled by
// log2(e) so softmax is a bare v_exp_f32:
//   score2 = 0.125*(0.5*sum log2(s) - 32 - 0.25*(La2+Lc2))
//          + 0.125*0.25*log2(e)*sum dmu^2/(s-eps)
// k-tiles staged to LDS by the Tensor Data Mover (hardware-padded rows).
// ---------------------------------------------------------------------------
__global__ __launch_bounds__(256) void dist_softmax(
    const float* __restrict__ muq, const float* __restrict__ muk,
    const float* __restrict__ sq, const float* __restrict__ sk,
    const float* __restrict__ Lab, const float* __restrict__ Lcb,
    float* __restrict__ attn) {
  __shared__ float aq[16][68], mq[16][68];
  __shared__ float ck[32][68], mk[32][68];
  __shared__ float row[16][516];
  __shared__ float la[16], lc[32];

  const int tid = threadIdx.x;
  const int wave = tid >> 5, lane = tid & 31;
  const int q0 = blockIdx.x * 16;
  const int h = blockIdx.y;

  for (int idx = tid; idx < 1024; idx += 256) {
    const int q = idx >> 6, d = idx & 63;
    const size_t g = (size_t)(q0 + q) * DM + h * DK + d;
    aq[q][d] = sq[g];
    mq[q][d] = muq[g];
  }
  if (tid < 16) la[tid] = Lab[(q0 + tid) * NH + h];

  const int kend = q0 + 16;          // exclusive bound on valid k (causal)
  const int q = tid & 15;
  const int kslot = tid >> 4;

  for (int chunk = 0; chunk < kend; chunk += 32) {
    __syncthreads();
#ifdef USE_TDM
    if (wave == 0) {
      tdm_load_tile_32x64(sk  + (size_t)chunk * DM + h * DK, &ck[0][0]);
      tdm_load_tile_32x64(muk + (size_t)chunk * DM + h * DK, &mk[0][0]);
      __builtin_amdgcn_s_wait_tensorcnt(0);
    }
#else
    for (int idx = tid; idx < 2048; idx += 256) {
      const int kl = idx >> 6, d = idx & 63;
      const size_t g = (size_t)(chunk + kl) * DM + h * DK + d;
      ck[kl][d] = sk[g];
      mk[kl][d] = muk[g];
    }
#endif
    if (tid < 32) lc[tid] = Lcb[(chunk + tid) * NH + h];
    __syncthreads();

#pragma unroll
    for (int i = 0; i < 2; ++i) {
      const int kl = kslot + 16 * i;
      const int kg = chunk + kl;
      if (kg > q0 + q) {
        row[q][kg] = -1e30f;          // causal mask
      } else {
        float acc1 = 0.f, acc2 = 0.f;
#pragma unroll 8
        for (int d = 0; d < DK; d += 2) {
          const v2f a2 = *(const v2f*)&aq[q][d];
          const v2f c2 = *(const v2f*)&ck[kl][d];
          const v2f m2 = *(const v2f*)&mq[q][d];
          const v2f k2 = *(const v2f*)&mk[kl][d];
          const float s0 = a2.x + c2.x, s1 = a2.y + c2.y;
          const float d0 = m2.x - k2.x, d1 = m2.y - k2.y;
          acc1 += __log2f(s0) + __log2f(s1);
          acc2 += d0 * d0 * __builtin_amdgcn_rcpf(s0 - EPSF)
                + d1 * d1 * __builtin_amdgcn_rcpf(s1 - EPSF);
        }
        const float t1 = 0.5f * acc1 - 32.f - 0.25f * (la[q] + lc[kl]);
        row[q][kg] = 0.125f * t1 + 0.04508422f * acc2;  // log2-domain score
      }
    }
  }
  __syncthreads();

  // softmax (base 2): wave w handles rows 2w, 2w+1 over valid prefix
#pragma unroll
  for (int rr = 0; rr < 2; ++rr) {
    const int r = 2 * wave + rr;
    const int qg = q0 + r;
    const int L = qg + 1;
    float m = -1e30f;
    for (int k = lane; k < L; k += 32) m = fmaxf(m, row[r][k]);
#pragma unroll
    for (int off = 16; off >= 1; off >>= 1) m = fmaxf(m, __shfl_xor(m, off, 32));
    float sum = 0.f;
    for (int k = lane; k < L; k += 32) sum += exp2f(row[r][k] - m);
#pragma unroll
    for (int off = 16; off >= 1; off >>= 1) sum += __shfl_xor(sum, off, 32);
    const float inv = __builtin_amdgcn_rcpf(sum);
    float* orow = attn + ((size_t)h * S + qg) * S;
    for (int k = lane; k < S; k += 32)
      orow[k] = (k < L) ? exp2f(row[r][k] - m) * inv : 0.f;
  }
}

// ---------------------------------------------------------------------------
// Kernel D: out_mu = attn @ mu_v, out_var = attn @ sg_v^2, fused fp32 WMMA.
// Per head: M=512, N=64 (4 tiles), K truncated to causal extent m0+16.
// ---------------------------------------------------------------------------
__global__ __launch_bounds__(128) void av_gemm(
    const float* __restrict__ attn, const float* __restrict__ mv,
    const float* __restrict__ vv, float* __restrict__ out) {
  const int tid = threadIdx.x;
  const int wave = tid >> 5, lane = tid & 31;
  const int lane16 = lane & 15, hi = lane >> 4;
  const int h = blockIdx.y;
  const int m0 = blockIdx.x * 64 + wave * 16;

  const float* A = attn + (size_t)h * S * S + (size_t)(m0 + lane16) * S + hi * 2;
  const int colbase = h * DK;

  v8f zero = {0.f,0.f,0.f,0.f,0.f,0.f,0.f,0.f};
  v8f am[4] = {zero, zero, zero, zero};
  v8f av[4] = {zero, zero, zero, zero};

  const int kend = m0 + 16;
  for (int k0 = 0; k0 < kend; k0 += 4) {
    const int kk = k0 + hi * 2;
    v2f a = *(const v2f*)(A + k0);
#pragma unroll
    for (int t = 0; t < 4; ++t) {
      const size_t b = (size_t)kk * DM + colbase + t * 16 + lane16;
      v2f bm; bm.x = mv[b]; bm.y = mv[b + DM];
      v2f bv; bv.x = vv[b]; bv.y = vv[b + DM];
      am[t] = __builtin_amdgcn_wmma_f32_16x16x4_f32(false, a, false, bm, (short)0, am[t], false, false);
      av[t] = __builtin_amdgcn_wmma_f32_16x16x4_f32(false, a, false, bv, (short)0, av[t], false, false);
    }
  }

#pragma unroll
  for (int t = 0; t < 4; ++t) {
    const int n = t * 16 + lane16;
#pragma unroll
    for (int r = 0; r < 8; ++r) {
      const int m = m0 + r + hi * 8;
      const size_t idx = ((size_t)h * S + m) * DK + n;
      out[idx] = am[t][r];
      float var = fmaxf(av[t][r], 0.f);
      out[(size_t)S * NH * DK + idx] = sqrtf(var + EPSF);
    }
  }
}

// ---------------------------------------------------------------------------
// Host side
// ---------------------------------------------------------------------------
extern "C" void kernel_launch(void* const* d_in, const int* in_sizes, int n_in,
                              void* d_out, int out_size, void* d_ws, size_t ws_size,
                              hipStream_t stream) {
  (void)in_sizes; (void)n_in; (void)out_size; (void)ws_size;
  const float* q   = (const float*)d_in[0];
  const float* k   = (const float*)d_in[1];
  const float* v   = (const float*)d_in[2];
  const float* qs  = (const float*)d_in[3];
  const float* ks  = (const float*)d_in[4];
  const float* vs  = (const float*)d_in[5];
  const float* wqm = (const float*)d_in[6];  const float* bqm = (const float*)d_in[7];
  const float* wkm = (const float*)d_in[8];  const float* bkm = (const float*)d_in[9];
  const float* wvm = (const float*)d_in[10]; const float* bvm = (const float*)d_in[11];
  const float* wqs = (const float*)d_in[12]; const float* bqs = (const float*)d_in[13];
  const float* wks = (const float*)d_in[14]; const float* bks = (const float*)d_in[15];
  const float* wvs = (const float*)d_in[16]; const float* bvs = (const float*)d_in[17];

  float* W = (float*)d_ws;
  const size_t P = (size_t)S * DM;            // 262144
  float* MU_Q = W;            float* MU_K = W + P;     float* MU_V = W + 2 * P;
  float* SQ   = W + 3 * P;    float* SK   = W + 4 * P; float* SV   = W + 5 * P;
  float* LA   = W + 6 * P;
  float* LC   = LA + (size_t)S * NH;
  float* ATT  = LC + (size_t)S * NH;          // 8*512*512

  dim3 gP(8, 4);
  proj_gemm<<<gP, 256, 0, stream>>>(q,  wqm, bqm, MU_Q);
  proj_gemm<<<gP, 256, 0, stream>>>(k,  wkm, bkm, MU_K);
  proj_gemm<<<gP, 256, 0, stream>>>(v,  wvm, bvm, MU_V);
  proj_gemm<<<gP, 256, 0, stream>>>(qs, wqs, bqs, SQ);
  proj_gemm<<<gP, 256, 0, stream>>>(ks, wks, bks, SK);
  proj_gemm<<<gP, 256, 0, stream>>>(vs, wvs, bvs, SV);

  sigma_post<<<512, 256, 0, stream>>>(SQ, LA, 1);
  sigma_post<<<512, 256, 0, stream>>>(SK, LC, 1);
  sigma_post<<<512, 256, 0, stream>>>(SV, nullptr, 0);

  dist_softmax<<<dim3(32, 8), 256, 0, stream>>>(MU_Q, MU_K, SQ, SK, LA, LC, ATT);

  av_gemm<<<dim3(8, 8), 128, 0, stream>>>(ATT, MU_V, SV, (float*)d_out);
}